// geo_transform_loss_70686571758255
// MI455X (gfx1250) — compile-verified
//
#include <hip/hip_runtime.h>
#include <math.h>

#define BS   256
#define NPTS 16384

typedef __attribute__((ext_vector_type(2))) float v2f;
typedef __attribute__((ext_vector_type(8))) float v8f;

// Per-batch constants (36 floats). Eight WMMA B-columns, each an affine form:
// value(p) = p . (vx,vy,vz) + offset
//   col0: py - py_re                  (sum |.|)
//   col1: px - px_re                  (sum |.|)
//   col2..4: half_c - pts_r_c         (min |.| -> m_plus)
//   col5..7: half_c + pts_r_c         (min |.| -> m_minus)
struct BatchConst {
    float col[8][4];
    float mask0;      // 1.0 if sym[0] != 1 else 0.0
    float flags[3];   // x,y,z symmetry flags = sym[3],sym[2],sym[1] > 0
};

__device__ __forceinline__ void cross3(const float a[3], const float b[3], float o[3]) {
    o[0] = a[1]*b[2] - a[2]*b[1];
    o[1] = a[2]*b[0] - a[0]*b[2];
    o[2] = a[0]*b[1] - a[1]*b[0];
}
__device__ __forceinline__ float dot3(const float a[3], const float b[3]) {
    return a[0]*b[0] + a[1]*b[1] + a[2]*b[2];
}
__device__ __forceinline__ void norm3(const float a[3], float o[3]) {
    float n = sqrtf(dot3(a, a)) + 1e-8f;
    o[0] = a[0]/n; o[1] = a[1]/n; o[2] = a[2]/n;
}
// Rodrigues rotation applied to a vector: R(k,th) v = c v + s (k x v) + (1-c) k (k.v)
__device__ __forceinline__ void rodrigues_apply(const float k[3], float th,
                                                const float v[3], float o[3]) {
    float c = cosf(th), s = sinf(th);
    float kxv[3]; cross3(k, v, kxv);
    float kd = dot3(k, v);
    for (int i = 0; i < 3; ++i)
        o[i] = c*v[i] + s*kxv[i] + (1.0f - c)*k[i]*kd;
}

// ---------------- Kernel 1: per-batch constants (tiny) ----------------
__global__ void geo_precompute_kernel(const float* __restrict__ rot1,
                                      const float* __restrict__ rot2,
                                      const float* __restrict__ rot1_f,
                                      const float* __restrict__ rot2_f,
                                      const float* __restrict__ tran,
                                      const float* __restrict__ size3,
                                      const float* __restrict__ mean_shape,
                                      const float* __restrict__ gR,
                                      const float* __restrict__ gT,
                                      const int*   __restrict__ sym,
                                      BatchConst*  __restrict__ bc) {
    int b = blockIdx.x * blockDim.x + threadIdx.x;
    if (b >= BS) return;

    float y[3] = { rot1[3*b], rot1[3*b+1], rot1[3*b+2] };
    float z[3] = { rot2[3*b], rot2[3*b+1], rot2[3*b+2] };
    float t[3] = { tran[3*b], tran[3*b+1], tran[3*b+2] };
    float gt[3]= { gT[3*b],   gT[3*b+1],   gT[3*b+2]   };

    // ---- face-loss rotation R via split Rodrigues ----
    float yxz[3]; cross3(y, z, yxz);
    float axis[3]; norm3(yxz, axis);
    float dyz = fminf(fmaxf(dot3(y, z), -1.0f), 1.0f);
    float theta = acosf(dyz);
    float c1 = rot1_f[b], c2 = rot2_f[b];
    float d = theta - 1.5707963267948966f;
    float th1 = c2 / (c1 + c2) * d;
    float th2 = c1 / (c1 + c2) * d;
    float new_y[3], new_x[3];
    rodrigues_apply(axis,  th1, y, new_y);
    rodrigues_apply(axis, -th2, z, new_x);
    float yn[3]; norm3(new_y, yn);
    float nxy[3]; cross3(new_x, yn, nxy);
    float zc[3]; norm3(nxy, zc);
    float xc[3]; cross3(yn, zc, xc);
    // R = stack([xc, yn, zc], axis=-1): column 0=xc, 1=yn, 2=zc

    // gR flat: gR[b*9 + row*3 + colIdx]; column i = (gR[0,i], gR[1,i], gR[2,i])
    const float* g = gR + b*9;
    float gc0[3] = { g[0], g[3], g[6] };  // gR[:,0]
    float gc1[3] = { g[1], g[4], g[7] };  // gR[:,1]

    BatchConst C;
    // col0: py - py_re = p.(rot1 - gR[:,1]) + (gT.gR[:,1] - tran.rot1)
    for (int i = 0; i < 3; ++i) C.col[0][i] = y[i] - gc1[i];
    C.col[0][3] = dot3(gt, gc1) - dot3(t, y);
    // col1: px - px_re = p.(rot2 - gR[:,0]) + (gT.gR[:,0] - tran.rot2)
    for (int i = 0; i < 3; ++i) C.col[1][i] = z[i] - gc0[i];
    C.col[1][3] = dot3(gt, gc0) - dot3(t, z);
    // cols 2..7: pts_r_c = p.R[:,c] - tran.R[:,c]
    //   col 2+c: half_c - pts_r_c  ->  (-R[:,c],  half_c + tran.R[:,c])
    //   col 5+c: half_c + pts_r_c  ->  (+R[:,c],  half_c - tran.R[:,c])
    const float* Rc[3] = { xc, yn, zc };
    for (int c = 0; c < 3; ++c) {
        float off = dot3(t, Rc[c]);
        float h   = 0.5f * (size3[3*b + c] + mean_shape[3*b + c]);
        for (int i = 0; i < 3; ++i) {
            C.col[2 + c][i] = -Rc[c][i];
            C.col[5 + c][i] =  Rc[c][i];
        }
        C.col[2 + c][3] = h + off;
        C.col[5 + c][3] = h - off;
    }
    C.mask0    = (sym[4*b + 0] != 1) ? 1.0f : 0.0f;
    C.flags[0] = (sym[4*b + 3] > 0) ? 1.0f : 0.0f;
    C.flags[1] = (sym[4*b + 2] > 0) ? 1.0f : 0.0f;
    C.flags[2] = (sym[4*b + 1] > 0) ? 1.0f : 0.0f;
    bc[b] = C;
}

// ---------------- Kernel 2: fused streaming pass over points ----------------
// One block per batch, 8 waves; each wave handles 16 points per WMMA.
// The chunk loop index is scalarized via readfirstlane so addresses are
// SGPR-base + constant per-lane voffset and loop control is pure SALU
// (EXEC all-1s at every WMMA by construction).
__global__ __launch_bounds__(256) void geo_main_kernel(
        const float* __restrict__ points,
        const BatchConst* __restrict__ bc,
        float* __restrict__ partial) {  // partial[b*8]: sum_y,sum_x,mp[3],mm[3]
    const int b    = blockIdx.x;
    const int tid  = threadIdx.x;
    const int lane = tid & 31;
    const int wave = __builtin_amdgcn_readfirstlane(tid >> 5);  // scalar 0..7
    const int col  = lane & 15;          // D column held by this lane
    const int hi   = lane >> 4;          // 0: K=0,1 / M=0..7 ; 1: K=2,3 / M=8..15

    const float* pts = points + (size_t)b * NPTS * 3;
    const BatchConst& C = bc[b];

    // Loop-invariant B operand: B is 4x16; VGPR0 holds rows 0/2 across lane
    // halves, VGPR1 rows 1/3 (mirrors the documented A layout).
    v2f Bv; Bv[0] = 0.0f; Bv[1] = 0.0f;
    if (col < 8) {
        const float* cv = C.col[col];
        Bv[0] = hi ? cv[2] : cv[0];      // row 0 (vx) / row 2 (vz)
        Bv[1] = hi ? cv[3] : cv[1];      // row 1 (vy) / row 3 (offset)
    }

    // Per-lane dword indices into a 48-dword (16-point) chunk.
    const int idx0 = col * 3 + (hi ? 2 : 0);   // K0 source: x (lo) / z (hi)
    const int idx1 = col * 3 + 1;              // K1 source for lo lanes: y

    float s0 = 0.0f, s1 = 0.0f;                       // sum chains
    float m0 = 3.402823466e38f, m1 = 3.402823466e38f; // min chains

#pragma unroll 2
    for (int chunk = wave; chunk < NPTS / 16; chunk += 8) {
        const float* pb = pts + chunk * 48;   // scalar (SGPR) base
        float a0  = pb[idx0];
        float a1l = pb[idx1];
        float a1  = hi ? 1.0f : a1l;          // homogeneous 1 for K=3
        v2f Av; Av[0] = a0; Av[1] = a1;
        v8f Cz = {};
        v8f D = __builtin_amdgcn_wmma_f32_16x16x4_f32(
            false, Av, false, Bv, (short)0, Cz, false, false);
#pragma unroll
        for (int r = 0; r < 8; r += 2) {
            float av0 = fabsf(D[r]);
            float av1 = fabsf(D[r + 1]);
            s0 += av0;             s1 += av1;
            m0 = fminf(m0, av0);   m1 = fminf(m1, av1);
        }
    }
    float accs = s0 + s1;
    float accm = fminf(m0, m1);

    // ---- block reduction: ds_add_f32 for sums, uint atomicMin for mins ----
    __shared__ float    s_sum[2];
    __shared__ unsigned s_min[6];
    if (tid == 0) {
        s_sum[0] = 0.0f; s_sum[1] = 0.0f;
        for (int i = 0; i < 6; ++i) s_min[i] = 0x7f7fffffu; // +FLT_MAX bits
    }
    __syncthreads();
    if (col < 2) atomicAdd(&s_sum[col], accs);
    if (col >= 2 && col < 8) {  // values >= 0 => IEEE order == uint order
        atomicMin(&s_min[col - 2], __float_as_uint(accm));
    }
    __syncthreads();
    if (tid == 0) {
        float* o = partial + b * 8;
        o[0] = s_sum[0];
        o[1] = s_sum[1];
        for (int i = 0; i < 6; ++i) o[2 + i] = __uint_as_float(s_min[i]);
    }
}

// ---------------- Kernel 3: combine across batches ----------------
__global__ __launch_bounds__(256) void geo_final_kernel(
        const float* __restrict__ partial,
        const BatchConst* __restrict__ bc,
        float* __restrict__ out) {
    __shared__ float r_y[BS], r_x[BS], r_cnt[BS], r_face[BS];
    const int b = threadIdx.x;
    const float* p = partial + b * 8;
    const BatchConst& C = bc[b];

    float mask = C.mask0;
    float face = 0.0f;
#pragma unroll
    for (int c = 0; c < 3; ++c) {
        float mpv = p[2 + c], mmv = p[5 + c];
        float mb  = fminf(mpv, mmv);
        float fl  = C.flags[c];
        float rp  = (fl > 0.0f) ? mb : mpv;
        float rm  = (fl > 0.0f) ? mb : mmv;
        face += rp + rm;
    }
    r_y[b]    = p[0];
    r_x[b]    = mask * (p[1] / (float)NPTS);
    r_cnt[b]  = mask;
    r_face[b] = face;
    __syncthreads();
    for (int s = BS / 2; s > 0; s >>= 1) {
        if (b < s) {
            r_y[b] += r_y[b + s]; r_x[b] += r_x[b + s];
            r_cnt[b] += r_cnt[b + s]; r_face[b] += r_face[b + s];
        }
        __syncthreads();
    }
    if (b == 0) {
        float res_y = r_y[0] / ((float)BS * (float)NPTS);
        float valid = r_cnt[0];
        float res_x = (valid > 0.0f) ? (r_x[0] / valid) : 0.0f;
        out[0] = res_y + res_x;                  // geo_point (GEO_P_W = 1)
        out[1] = (r_face[0] / (float)BS) / 6.0f; // geo_face  (GEO_F_W = 1)
    }
}

extern "C" void kernel_launch(void* const* d_in, const int* in_sizes, int n_in,
                              void* d_out, int out_size, void* d_ws, size_t ws_size,
                              hipStream_t stream) {
    const float* points     = (const float*)d_in[0];
    const float* rot1       = (const float*)d_in[1];
    const float* rot2       = (const float*)d_in[2];
    const float* rot1_f     = (const float*)d_in[3];
    const float* rot2_f     = (const float*)d_in[4];
    const float* tran       = (const float*)d_in[5];
    const float* size3      = (const float*)d_in[6];
    const float* mean_shape = (const float*)d_in[7];
    const float* gR         = (const float*)d_in[8];
    const float* gT         = (const float*)d_in[9];
    const int*   sym        = (const int*)d_in[10];

    BatchConst* bc = (BatchConst*)d_ws;
    float* partial = (float*)((char*)d_ws + ((BS * sizeof(BatchConst) + 255) & ~255));
    float* out = (float*)d_out;

    geo_precompute_kernel<<<1, BS, 0, stream>>>(rot1, rot2, rot1_f, rot2_f, tran,
                                                size3, mean_shape, gR, gT, sym, bc);
    geo_main_kernel<<<BS, 256, 0, stream>>>(points, bc, partial);
    geo_final_kernel<<<1, BS, 0, stream>>>(partial, bc, out);
}